// AttnDecoderGRU_8340826489406
// MI455X (gfx1250) — compile-verified
//
#include <hip/hip_runtime.h>
#include <math.h>

#define H_ 1024
#define V_ 50257
#define S_ 2048

typedef float v2f __attribute__((ext_vector_type(2)));
typedef float v8f __attribute__((ext_vector_type(8)));

// ---------------------------------------------------------------------------
// Generic wave-per-row GEMV: y[r] = dot(A[r,:], x) + b0[r] + b1[r]; act=1 -> relu
// One wave (32 lanes) per row, float4 loads, shuffle reduction.
// ---------------------------------------------------------------------------
__global__ void gemv_kernel(const float* __restrict__ A, const float* __restrict__ x,
                            const float* __restrict__ b0, const float* __restrict__ b1,
                            float* __restrict__ y, int rows, int cols, int act) {
    const int wave = threadIdx.x >> 5;
    const int lane = threadIdx.x & 31;
    const int row  = blockIdx.x * (blockDim.x >> 5) + wave;
    if (row >= rows) return;
    const float* a = A + (size_t)row * (size_t)cols;
    float sum = 0.0f;
    for (int k = lane * 4; k < cols; k += 32 * 4) {
        const float4 av = *(const float4*)(a + k);
        const float4 xv = *(const float4*)(x + k);
        sum += av.x * xv.x + av.y * xv.y + av.z * xv.z + av.w * xv.w;
    }
    for (int off = 16; off; off >>= 1) sum += __shfl_down(sum, off, 32);
    if (lane == 0) {
        float r = sum;
        if (b0) r += b0[row];
        if (b1) r += b1[row];
        if (act == 1) r = fmaxf(r, 0.0f);
        y[row] = r;
    }
}

// ---------------------------------------------------------------------------
// Attention scores via V_WMMA_F32_16X16X4_F32:
//   e[s] = sum_j va[j] * tanh( dot(enc[s,:], Ua_w[j,:]) + bias[j] )
// Block = 16-row strip of enc staged in LDS; 8 waves sweep 64 column tiles.
// A (16x4 f32): lane (0..15|16..31) -> M=lane&15; VGPR0=K0|K2, VGPR1=K1|K3.
// B (4x16 f32): lanes -> N=lane&15; same K split (B[k][n] = Ua_w[n][k]).
// C (16x16 f32): c[i] -> row = i + 8*(lane>=16), col = lane&15.
// ---------------------------------------------------------------------------
__global__ void scores_e_kernel(const float* __restrict__ enc, const float* __restrict__ Ua_w,
                                const float* __restrict__ bias, const float* __restrict__ va,
                                float* __restrict__ e_out) {
    __shared__ float ldsA[16 * H_];
    __shared__ float lds_e[16];
    const int tid = threadIdx.x;
    const int s0  = blockIdx.x * 16;

    if (tid < 16) lds_e[tid] = 0.0f;
    for (int idx = tid * 4; idx < 16 * H_; idx += 256 * 4) {
        *(float4*)(ldsA + idx) = *(const float4*)(enc + (size_t)s0 * H_ + idx);
    }
    __syncthreads();

    const int wave = tid >> 5;
    const int lane = tid & 31;
    const int half = lane >> 4;   // 0: lanes 0-15, 1: lanes 16-31
    const int l15  = lane & 15;

    for (int t = wave; t < H_ / 16; t += 8) {
        const int n0 = t * 16;
        v8f c = {0.f, 0.f, 0.f, 0.f, 0.f, 0.f, 0.f, 0.f};
        const float* arow = ldsA + l15 * H_ + 2 * half;                      // LDS
        const float* brow = Ua_w + (size_t)(n0 + l15) * H_ + 2 * half;      // L2-resident
        for (int k = 0; k < H_; k += 4) {
            const v2f a = *(const v2f*)(arow + k);
            const v2f b = *(const v2f*)(brow + k);
            c = __builtin_amdgcn_wmma_f32_16x16x4_f32(false, a, false, b,
                                                      (short)0, c, false, false);
        }
        // fused epilogue: tanh + weight by va, then reduce over the 16 columns
        const int   cg = n0 + l15;
        const float bs = bias[cg];
        const float vv = va[cg];
        float p[8];
#pragma unroll
        for (int i = 0; i < 8; ++i) p[i] = vv * tanhf(c[i] + bs);
#pragma unroll
        for (int i = 0; i < 8; ++i) {
            float xsum = p[i];
            for (int off = 8; off; off >>= 1) xsum += __shfl_xor(xsum, off, 16);
            if (l15 == 0) atomicAdd(&lds_e[i + 8 * half], xsum);
        }
    }
    __syncthreads();
    if (tid < 16) e_out[s0 + tid] = lds_e[tid];
}

// softmax over S (single block)
__global__ void softmax_s_kernel(const float* __restrict__ e, float* __restrict__ w_ws,
                                 float* __restrict__ w_out) {
    __shared__ float red[256];
    __shared__ float s_max, s_sum;
    const int tid = threadIdx.x;
    float m = -INFINITY;
    for (int i = tid; i < S_; i += 256) m = fmaxf(m, e[i]);
    red[tid] = m; __syncthreads();
    for (int o = 128; o; o >>= 1) { if (tid < o) red[tid] = fmaxf(red[tid], red[tid + o]); __syncthreads(); }
    if (tid == 0) s_max = red[0];
    __syncthreads();
    float sum = 0.0f;
    for (int i = tid; i < S_; i += 256) sum += expf(e[i] - s_max);
    red[tid] = sum; __syncthreads();
    for (int o = 128; o; o >>= 1) { if (tid < o) red[tid] += red[tid + o]; __syncthreads(); }
    if (tid == 0) s_sum = red[0];
    __syncthreads();
    const float inv = 1.0f / s_sum;
    for (int i = tid; i < S_; i += 256) {
        const float w = expf(e[i] - s_max) * inv;
        w_ws[i]  = w;
        w_out[i] = w;
    }
}

// context[h] = sum_s attn[s] * enc[s][h]   (coalesced column streaming)
__global__ void context_kernel(const float* __restrict__ enc, const float* __restrict__ attn,
                               float* __restrict__ ctx) {
    const int h = blockIdx.x * blockDim.x + threadIdx.x;
    float sum = 0.0f;
    for (int s = 0; s < S_; ++s) sum += attn[s] * enc[(size_t)s * H_ + h];
    ctx[h] = sum;
}

// cat_in = [emb[input], context]
__global__ void build_cat_in_kernel(const int* __restrict__ input, const float* __restrict__ emb,
                                    const float* __restrict__ ctx, float* __restrict__ cat_in) {
    const int j = blockIdx.x * blockDim.x + threadIdx.x;
    if (j < H_) cat_in[j] = emb[(size_t)input[0] * H_ + j];
    else        cat_in[j] = ctx[j - H_];
}

// GRU gate fusion + build cat2 = [context, h_new]; writes hidden_out
__global__ void gru_gate_kernel(const float* __restrict__ gi, const float* __restrict__ gh,
                                const float* __restrict__ hidden, const float* __restrict__ ctx,
                                float* __restrict__ hid_out, float* __restrict__ cat2) {
    const int j = blockIdx.x * blockDim.x + threadIdx.x;
    const float r = 1.0f / (1.0f + expf(-(gi[j] + gh[j])));
    const float z = 1.0f / (1.0f + expf(-(gi[H_ + j] + gh[H_ + j])));
    const float n = tanhf(gi[2 * H_ + j] + r * gh[2 * H_ + j]);
    const float h0 = hidden[j];
    const float hn = (1.0f - z) * n + z * h0;
    hid_out[j]   = hn;
    cat2[j]      = ctx[j];
    cat2[H_ + j] = hn;
}

// log_softmax over V (single block, three streaming passes over 0.2 MB)
__global__ void log_softmax_v_kernel(const float* __restrict__ logits, float* __restrict__ out) {
    __shared__ float red[1024];
    __shared__ float s_max, s_log;
    const int tid = threadIdx.x;
    float m = -INFINITY;
    for (int i = tid; i < V_; i += 1024) m = fmaxf(m, logits[i]);
    red[tid] = m; __syncthreads();
    for (int o = 512; o; o >>= 1) { if (tid < o) red[tid] = fmaxf(red[tid], red[tid + o]); __syncthreads(); }
    if (tid == 0) s_max = red[0];
    __syncthreads();
    float sum = 0.0f;
    for (int i = tid; i < V_; i += 1024) sum += expf(logits[i] - s_max);
    red[tid] = sum; __syncthreads();
    for (int o = 512; o; o >>= 1) { if (tid < o) red[tid] += red[tid + o]; __syncthreads(); }
    if (tid == 0) s_log = logf(red[0]);
    __syncthreads();
    for (int i = tid; i < V_; i += 1024) out[i] = logits[i] - s_max - s_log;
}

extern "C" void kernel_launch(void* const* d_in, const int* in_sizes, int n_in,
                              void* d_out, int out_size, void* d_ws, size_t ws_size,
                              hipStream_t stream) {
    (void)in_sizes; (void)n_in; (void)out_size; (void)ws_size;
    const int*   input    = (const int*)  d_in[0];
    const float* hidden   = (const float*)d_in[1];
    const float* enc      = (const float*)d_in[2];
    const float* emb      = (const float*)d_in[3];
    const float* Ua_w     = (const float*)d_in[4];
    const float* Ua_b     = (const float*)d_in[5];
    const float* Wa_w     = (const float*)d_in[6];
    const float* Wa_b     = (const float*)d_in[7];
    const float* va       = (const float*)d_in[8];
    const float* comb_w   = (const float*)d_in[9];
    const float* comb_b   = (const float*)d_in[10];
    const float* gru_w_ih = (const float*)d_in[11];
    const float* gru_w_hh = (const float*)d_in[12];
    const float* gru_b_ih = (const float*)d_in[13];
    const float* gru_b_hh = (const float*)d_in[14];
    const float* out_w    = (const float*)d_in[15];
    const float* out_b    = (const float*)d_in[16];

    float* out       = (float*)d_out;          // [V_] log_probs | [H_] hidden | [S_] attn
    float* o_logp    = out;
    float* o_hidden  = out + V_;
    float* o_attn    = out + V_ + H_;

    float* ws      = (float*)d_ws;
    float* bias    = ws;                 // H_
    float* e       = bias   + H_;        // S_
    float* attn    = e      + S_;        // S_
    float* ctx     = attn   + S_;        // H_
    float* cat_in  = ctx    + H_;        // 2H_
    float* x_comb  = cat_in + 2 * H_;    // H_
    float* gi      = x_comb + H_;        // 3H_
    float* gh      = gi     + 3 * H_;    // 3H_
    float* cat2    = gh     + 3 * H_;    // 2H_
    float* logits  = cat2   + 2 * H_;    // V_

    // 1) bias = hidden @ Wa_w^T + Ua_b + Wa_b
    gemv_kernel<<<H_ / 8, 256, 0, stream>>>(Wa_w, hidden, Ua_b, Wa_b, bias, H_, H_, 0);
    // 2) WMMA attention scores -> e[s]
    scores_e_kernel<<<S_ / 16, 256, 0, stream>>>(enc, Ua_w, bias, va, e);
    // 3) softmax over S -> attn (ws + output slot)
    softmax_s_kernel<<<1, 256, 0, stream>>>(e, attn, o_attn);
    // 4) context = attn @ enc
    context_kernel<<<H_ / 256, 256, 0, stream>>>(enc, attn, ctx);
    // 5) cat_in = [emb[input], context]
    build_cat_in_kernel<<<(2 * H_) / 256, 256, 0, stream>>>(input, emb, ctx, cat_in);
    // 6) x = relu(cat_in @ comb_w^T + comb_b)
    gemv_kernel<<<H_ / 8, 256, 0, stream>>>(comb_w, cat_in, comb_b, nullptr, x_comb, H_, 2 * H_, 1);
    // 7) gi = x @ gru_w_ih^T + b_ih ; gh = h0 @ gru_w_hh^T + b_hh
    gemv_kernel<<<(3 * H_) / 8, 256, 0, stream>>>(gru_w_ih, x_comb, gru_b_ih, nullptr, gi, 3 * H_, H_, 0);
    gemv_kernel<<<(3 * H_) / 8, 256, 0, stream>>>(gru_w_hh, hidden, gru_b_hh, nullptr, gh, 3 * H_, H_, 0);
    // 8) GRU gates -> hidden_out, cat2
    gru_gate_kernel<<<H_ / 256, 256, 0, stream>>>(gi, gh, hidden, ctx, o_hidden, cat2);
    // 9) logits = cat2 @ out_w^T + out_b   (412 MB stream: the bandwidth floor)
    gemv_kernel<<<(V_ + 7) / 8, 256, 0, stream>>>(out_w, cat2, out_b, nullptr, logits, V_, 2 * H_, 0);
    // 10) log_softmax over V
    log_softmax_v_kernel<<<1, 1024, 0, stream>>>(logits, o_logp);
}